// OWAPoolingNew_87522843560663
// MI455X (gfx1250) — compile-verified
//
#include <hip/hip_runtime.h>
#include <stdint.h>

#define B_   16
#define H_   224
#define W_   224
#define C_   64
#define OH_  112
#define OW_  112
#define C4_  16      // C_/4 float4 groups per pixel

// Block tile: 2 output rows x 16 output cols x all 64 channels
#define TOH  2
#define TOW  16
#define TIH  (TOH * 2 + 1)   // 5 input rows
#define TIW  (TOW * 2 + 1)   // 33 input cols

__global__ __launch_bounds__(256)
void owa_pool_kernel(const float* __restrict__ in,
                     const float* __restrict__ wk,
                     float* __restrict__ out)
{
    __shared__ float4 tile[TIH * TIW * C4_];  // 5*33*16 float4 = 42,240 B
    __shared__ float4 wlds[9 * C4_];          // 144 float4   =  2,304 B

    const int tid = threadIdx.x;
    const int bw = blockIdx.x % (OW_ / TOW);
    const int bh = (blockIdx.x / (OW_ / TOW)) % (OH_ / TOH);
    const int b  = blockIdx.x / ((OW_ / TOW) * (OH_ / TOH));

    const int gr0 = bh * (TOH * 2);   // first input row of tile
    const int gc0 = bw * (TOW * 2);   // first input col of tile

    // ---- stage weights (9 x 64 fp32 = 144 float4) ----
    if (tid < 9 * C4_) {
        wlds[tid] = ((const float4*)wk)[tid];
    }

    // ---- stage input tile via CDNA5 async global->LDS DMA ----
    // In-bounds elements: GLOBAL_LOAD_ASYNC_TO_LDS_B128 (cache->LDS, no VGPR
    // round-trip, tracked by ASYNCcnt). Out-of-bounds pad elements: plain LDS
    // store of zero. Disjoint addresses -> no DScnt/ASYNCcnt ordering hazard.
    const float4* in4 = (const float4*)in;
    for (int i = tid; i < TIH * TIW * C4_; i += 256) {
        int r   = i / (TIW * C4_);
        int rem = i - r * (TIW * C4_);
        int col = rem / C4_;
        int c4  = rem - col * C4_;
        int gr = gr0 + r;
        int gc = gc0 + col;
        if (gr < H_ && gc < W_) {
            uint32_t ldsa = (uint32_t)(uintptr_t)(tile + i);   // low 32 bits of
                                                               // flat addr = LDS offset
            unsigned long long gpa = (unsigned long long)(uintptr_t)
                (in4 + ((size_t)((b * H_ + gr) * W_ + gc)) * C4_ + c4);
            asm volatile("global_load_async_to_lds_b128 %0, %1, off"
                         :: "v"(ldsa), "v"(gpa)
                         : "memory");
        } else {
            tile[i] = make_float4(0.f, 0.f, 0.f, 0.f);
        }
    }
    asm volatile("s_wait_asynccnt 0x0" ::: "memory");
    __syncthreads();

    const int c4 = tid & (C4_ - 1);   // channel group 0..15 (fastest -> coalesced)
    const int wl = tid >> 4;          // local output col 0..15

    #pragma unroll
    for (int hl = 0; hl < TOH; ++hl) {
        float4 v[9];
        #pragma unroll
        for (int kr = 0; kr < 3; ++kr)
            #pragma unroll
            for (int kc = 0; kc < 3; ++kc)
                v[kr * 3 + kc] =
                    tile[((hl * 2 + kr) * TIW + (wl * 2 + kc)) * C4_ + c4];

        // Optimal 25-comparator sorting network for 9 inputs (ascending),
        // per channel component; clang fuses chains into v_min3/v_max3/v_med3.
        #define F4CE(i,j) { float4 a_ = v[i], b_ = v[j];              \
            v[i].x = fminf(a_.x, b_.x); v[j].x = fmaxf(a_.x, b_.x);   \
            v[i].y = fminf(a_.y, b_.y); v[j].y = fmaxf(a_.y, b_.y);   \
            v[i].z = fminf(a_.z, b_.z); v[j].z = fmaxf(a_.z, b_.z);   \
            v[i].w = fminf(a_.w, b_.w); v[j].w = fmaxf(a_.w, b_.w); }
        F4CE(0,3) F4CE(1,7) F4CE(2,5) F4CE(4,8)
        F4CE(0,7) F4CE(2,4) F4CE(3,8) F4CE(5,6)
        F4CE(0,2) F4CE(1,3) F4CE(4,5) F4CE(7,8)
        F4CE(1,4) F4CE(3,6) F4CE(5,7)
        F4CE(0,1) F4CE(2,4) F4CE(3,5) F4CE(6,8)
        F4CE(2,3) F4CE(4,5) F4CE(6,7)
        F4CE(1,2) F4CE(3,4) F4CE(5,6)
        #undef F4CE

        // Descending-sort dot == ascending values dotted with reversed weights
        float4 acc = make_float4(0.f, 0.f, 0.f, 0.f);
        #pragma unroll
        for (int j = 0; j < 9; ++j) {
            float4 w = wlds[(8 - j) * C4_ + c4];
            acc.x = fmaf(v[j].x, w.x, acc.x);
            acc.y = fmaf(v[j].y, w.y, acc.y);
            acc.z = fmaf(v[j].z, w.z, acc.z);
            acc.w = fmaf(v[j].w, w.w, acc.w);
        }

        const int oh = bh * TOH + hl;
        const int ow = bw * TOW + wl;
        ((float4*)out)[((size_t)((b * OH_ + oh) * OW_ + ow)) * C4_ + c4] = acc;
    }
}

extern "C" void kernel_launch(void* const* d_in, const int* in_sizes, int n_in,
                              void* d_out, int out_size, void* d_ws, size_t ws_size,
                              hipStream_t stream) {
    const float* in = (const float*)d_in[0];   // (16,224,224,64) fp32
    const float* wk = (const float*)d_in[1];   // (9,64) fp32
    float* out = (float*)d_out;                // (16,112,112,64) fp32

    const int grid = B_ * (OH_ / TOH) * (OW_ / TOW);   // 16*56*7 = 6272 blocks
    owa_pool_kernel<<<grid, 256, 0, stream>>>(in, wk, out);
}